// ArapEigenEnergy_46059229282951
// MI455X (gfx1250) — compile-verified
//
#include <hip/hip_runtime.h>

typedef __attribute__((ext_vector_type(2))) float v2f;
typedef __attribute__((ext_vector_type(8))) float v8f;

#define NV 25000          // vertices
#define TN 75000          // 3*N
#define MM 512            // modes
#define BB 16             // batch
#define KK 16             // max neighbors
#define CHUNK 512         // i-elements per workgroup chunk
#define NCHUNK 147        // ceil(75000/512)
#define NWG_EIG 147       // one WG per chunk
#define LDSS 17           // padded LDS row stride (floats) for bank-conflict-free access
#define G2 64             // ARAP blocks

// ---------------------------------------------------------------------------
// Kernel 1: partial GEMMs  coeff[m,b] = eigVT * d  and  coeffV[b,m] = d * eigV
// using V_WMMA_F32_16X16X4_F32 (full fp32 matrix core path).
// ---------------------------------------------------------------------------
__launch_bounds__(256)
__global__ void eig_partial_kernel(const float* __restrict__ xyz1,
                                   const float* __restrict__ recon,
                                   const float* __restrict__ eigV,   // [TN][MM] row-major
                                   const float* __restrict__ eigVT,  // [MM][TN] row-major
                                   float* __restrict__ partA,        // [NWG][MM][BB]
                                   float* __restrict__ partB)        // [NWG][BB][MM]
{
    __shared__ float dsh[CHUNK * LDSS];

    const int wg   = blockIdx.x;
    const int tid  = threadIdx.x;
    const int wave = tid >> 5;
    const int lane = tid & 31;
    const int lo   = lane & 15;   // M/N/B index within tile
    const int hi   = lane >> 4;   // selects K block {0,1} vs {2,3}

    v8f accA[4], accB[4];
    v8f zero;
    #pragma unroll
    for (int r = 0; r < 8; ++r) zero[r] = 0.0f;
    #pragma unroll
    for (int t = 0; t < 4; ++t) { accA[t] = zero; accB[t] = zero; }

    for (int c = wg; c < NCHUNK; c += NWG_EIG) {
        const int i0 = c * CHUNK;
        __syncthreads();
        // Stage d[b][i] = recon[b,i] - xyz1[i] into LDS as [i][b] (stride 17).
        for (int idx = tid; idx < BB * CHUNK; idx += 256) {
            int b  = idx >> 9;         // / CHUNK
            int il = idx & (CHUNK - 1);
            int i  = i0 + il;
            int ic = (i < TN) ? i : (TN - 1);
            float v = recon[b * TN + ic] - xyz1[ic];
            if (i >= TN) v = 0.0f;     // zero tail -> OOB eig values multiply by 0
            dsh[il * LDSS + b] = v;
        }
        __syncthreads();

        for (int s = 0; s < CHUNK; s += 4) {
            // clamp global column base so the last chunk never reads OOB
            int iq  = i0 + s;
            int iqc = (iq > TN - 4) ? (TN - 4) : iq;

            // shared d fragment: serves as B (4k x 16b) for coeff and A (16b x 4k) for coeffV
            v2f df;
            df[0] = dsh[(s + 2 * hi) * LDSS + lo];
            df[1] = dsh[(s + 2 * hi + 1) * LDSS + lo];

            #pragma unroll
            for (int t = 0; t < 4; ++t) {
                const int m0 = wave * 64 + t * 16;

                // coeff: A = eigVT tile (16m x 4k), row (m0+lo), cols iqc+2*hi..+1
                const float* pA = eigVT + (size_t)(m0 + lo) * TN + iqc + 2 * hi;
                v2f af; af[0] = pA[0]; af[1] = pA[1];
                accA[t] = __builtin_amdgcn_wmma_f32_16x16x4_f32(
                    false, af, false, df, (short)0, accA[t], false, false);

                // coeffV: B = eigV tile (4k x 16m), rows iqc+2*hi..+1, col m0+lo
                const float* pB = eigV + (size_t)(iqc + 2 * hi) * MM + m0 + lo;
                v2f bf; bf[0] = pB[0]; bf[1] = pB[MM];
                accB[t] = __builtin_amdgcn_wmma_f32_16x16x4_f32(
                    false, df, false, bf, (short)0, accB[t], false, false);
            }
        }
    }

    // D layout (16x16 f32): VGPR r, lanes 0-15 -> row r, lanes 16-31 -> row r+8
    #pragma unroll
    for (int t = 0; t < 4; ++t) {
        const int m0 = wave * 64 + t * 16;
        #pragma unroll
        for (int r = 0; r < 8; ++r) {
            int mA = m0 + r + 8 * hi;              // coeff tile: D[m][b]
            partA[((size_t)wg * MM + mA) * BB + lo] = accA[t][r];
            int bB = r + 8 * hi;                   // coeffV tile: D[b][m]
            partB[(size_t)wg * (BB * MM) + bB * MM + m0 + lo] = accB[t][r];
        }
    }
}

// ---------------------------------------------------------------------------
// Kernel 2: ARAP edge energy block partials (deterministic fixed-order reduce)
// ---------------------------------------------------------------------------
__launch_bounds__(256)
__global__ void arap_kernel(const float* __restrict__ xyz1,
                            const float* __restrict__ wmat,
                            const float* __restrict__ recon,
                            const int* __restrict__ nbrs,
                            const int* __restrict__ nnb,
                            float* __restrict__ arap_part)   // [G2][BB]
{
    __shared__ float red[256 * BB];
    float acc[BB];
    #pragma unroll
    for (int b = 0; b < BB; ++b) acc[b] = 0.0f;

    for (int n = blockIdx.x * 256 + threadIdx.x; n < NV; n += G2 * 256) {
        const int nn = nnb[n];
        const float x0 = xyz1[n * 3 + 0];
        const float y0 = xyz1[n * 3 + 1];
        const float z0 = xyz1[n * 3 + 2];
        float rx[BB], ry[BB], rz[BB];
        #pragma unroll
        for (int b = 0; b < BB; ++b) {
            const float* p = recon + ((size_t)b * NV + n) * 3;
            rx[b] = p[0]; ry[b] = p[1]; rz[b] = p[2];
        }
        for (int k = 0; k < KK; ++k) {
            const float wk = (k < nn) ? wmat[n * KK + k] : 0.0f;
            const int j = nbrs[n * KK + k];
            const float ex = x0 - xyz1[j * 3 + 0];
            const float ey = y0 - xyz1[j * 3 + 1];
            const float ez = z0 - xyz1[j * 3 + 2];
            #pragma unroll
            for (int b = 0; b < BB; ++b) {
                const float* q = recon + ((size_t)b * NV + j) * 3;
                float dx = (rx[b] - q[0]) - ex;
                float dy = (ry[b] - q[1]) - ey;
                float dz = (rz[b] - q[2]) - ez;
                acc[b] += wk * (dx * dx + dy * dy + dz * dz);
            }
        }
    }
    #pragma unroll
    for (int b = 0; b < BB; ++b) red[threadIdx.x * BB + b] = acc[b];
    __syncthreads();
    if (threadIdx.x < BB) {
        float s = 0.0f;
        for (int j = 0; j < 256; ++j) s += red[j * BB + threadIdx.x];
        arap_part[blockIdx.x * BB + threadIdx.x] = s;
    }
}

// ---------------------------------------------------------------------------
// Kernel 3: combine per-WG GEMM partials into 0.5*eigC*coeff*coeffV (masked)
// ---------------------------------------------------------------------------
__launch_bounds__(256)
__global__ void combine_kernel(const float* __restrict__ partA,
                               const float* __restrict__ partB,
                               const float* __restrict__ eigC,
                               const int* __restrict__ nComp,
                               float* __restrict__ prod)     // [BB*MM]
{
    const int e = blockIdx.x * 256 + threadIdx.x;
    if (e >= BB * MM) return;
    const int b = e / MM;
    const int m = e - b * MM;
    float sA = 0.0f, sB = 0.0f;
    for (int wg = 0; wg < NWG_EIG; ++wg) {
        sA += partA[((size_t)wg * MM + m) * BB + b];
        sB += partB[(size_t)wg * (BB * MM) + b * MM + m];
    }
    prod[e] = (m >= nComp[0]) ? 0.5f * eigC[m] * sA * sB : 0.0f;
}

// ---------------------------------------------------------------------------
// Kernel 4: final deterministic reduction -> mean over batch
// ---------------------------------------------------------------------------
__launch_bounds__(256)
__global__ void final_kernel(const float* __restrict__ prod,
                             const float* __restrict__ arap_part,
                             float* __restrict__ out)
{
    __shared__ float red[256];
    float s = 0.0f;
    for (int j = threadIdx.x; j < BB * MM; j += 256) s += prod[j];
    for (int j = threadIdx.x; j < G2 * BB; j += 256) s += arap_part[j];
    red[threadIdx.x] = s;
    __syncthreads();
    for (int off = 128; off > 0; off >>= 1) {
        if (threadIdx.x < off) red[threadIdx.x] += red[threadIdx.x + off];
        __syncthreads();
    }
    if (threadIdx.x == 0) out[0] = red[0] * (1.0f / BB);
}

extern "C" void kernel_launch(void* const* d_in, const int* in_sizes, int n_in,
                              void* d_out, int out_size, void* d_ws, size_t ws_size,
                              hipStream_t stream) {
    (void)in_sizes; (void)n_in; (void)out_size; (void)ws_size;
    const float* xyz1  = (const float*)d_in[0];
    const float* wmat  = (const float*)d_in[1];
    const float* recon = (const float*)d_in[2];
    const float* eigC  = (const float*)d_in[3];
    const float* eigV  = (const float*)d_in[4];
    const float* eigVT = (const float*)d_in[5];
    const int*   nbrs  = (const int*)d_in[6];
    const int*   nnb   = (const int*)d_in[7];
    const int*   nComp = (const int*)d_in[8];
    float* out = (float*)d_out;

    float* ws        = (float*)d_ws;
    float* partA     = ws;                                   // NWG*MM*BB
    float* partB     = partA + (size_t)NWG_EIG * MM * BB;    // NWG*BB*MM
    float* arap_part = partB + (size_t)NWG_EIG * MM * BB;    // G2*BB
    float* prod      = arap_part + (size_t)G2 * BB;          // BB*MM

    eig_partial_kernel<<<NWG_EIG, 256, 0, stream>>>(xyz1, recon, eigV, eigVT, partA, partB);
    arap_kernel<<<G2, 256, 0, stream>>>(xyz1, wmat, recon, nbrs, nnb, arap_part);
    combine_kernel<<<(BB * MM + 255) / 256, 256, 0, stream>>>(partA, partB, eigC, nComp, prod);
    final_kernel<<<1, 256, 0, stream>>>(prod, arap_part, out);
}